// GlobalAttention_73710228734486
// MI455X (gfx1250) — compile-verified
//
#include <hip/hip_runtime.h>
#include <hip/hip_bf16.h>

#define NN      262144
#define HID     512
#define NG      64
#define NGRAPH  1024
#define KC      64
#define KCP     (KC + 2)   // padded k-extent for transposed W tile (bank decorrelation)
#define EPSF    1e-16f

typedef __attribute__((ext_vector_type(2))) float v2f;
typedef __attribute__((ext_vector_type(8))) float v8f;

// ---------------- segment boundary setup ----------------
__global__ __launch_bounds__(256) void init_seg_kernel(int* seg_start, int* seg_end) {
    int g = blockIdx.x * 256 + threadIdx.x;
    if (g < NGRAPH) { seg_start[g] = 0; seg_end[g] = 0; }
}

__global__ __launch_bounds__(256) void seg_bounds_kernel(const int* __restrict__ batch,
                                                         int* seg_start, int* seg_end) {
    int i = blockIdx.x * 256 + threadIdx.x;
    if (i >= NN) return;
    int b = batch[i];
    if (i == 0      || batch[i - 1] != b) seg_start[b] = i;
    if (i == NN - 1 || batch[i + 1] != b) seg_end[b]   = i + 1;
}

// ---------------- gate = x @ W via v_wmma_f32_16x16x4_f32 ----------------
// Block: 128 threads = 4 waves. Each wave: 32 rows (two 16-row M tiles) x 64 gate cols.
// Block covers 128 rows; grid = NN/128 = 2048 blocks.
__global__ __launch_bounds__(128) void gate_gemm_kernel(const float* __restrict__ x,
                                                        const float* __restrict__ W,
                                                        float* __restrict__ gate) {
    __shared__ float sWt[NG][KCP];   // transposed W K-chunk: sWt[gateCol][k]

    const int tid  = threadIdx.x;
    const int lane = tid & 31;
    const int wave = tid >> 5;
    const int rowBase = blockIdx.x * 128 + wave * 32;

    // accumulators: 8 x v8f = 64 VGPRs
    v8f acc00 = {}, acc01 = {}, acc02 = {}, acc03 = {};   // rows rowBase .. +15
    v8f acc10 = {}, acc11 = {}, acc12 = {}, acc13 = {};   // rows rowBase+16 .. +31

    const int koff = (lane >> 4) * 2;   // K sub-offset per lane half
    const int c    = lane & 15;         // column within a 16-wide tile
    const float* xr0 = x + (size_t)(rowBase + c) * HID + koff;
    const float* xr1 = xr0 + (size_t)16 * HID;

    // W-chunk fill mapping: thread t -> k-row (t>>1), cols (t&1)*32 .. +31
    const int frow  = tid >> 1;
    const int fcol0 = (tid & 1) * 32;

    for (int kc = 0; kc < HID; kc += KC) {
        // Coalesced global read of W[kc+frow][fcol0..fcol0+31], transposed into LDS
        const float* wrow = W + (size_t)(kc + frow) * NG + fcol0;
        #pragma unroll
        for (int j = 0; j < 8; ++j) {
            float4 v = *(const float4*)(wrow + j * 4);
            sWt[fcol0 + j * 4 + 0][frow] = v.x;
            sWt[fcol0 + j * 4 + 1][frow] = v.y;
            sWt[fcol0 + j * 4 + 2][frow] = v.z;
            sWt[fcol0 + j * 4 + 3][frow] = v.w;
        }
        __syncthreads();

        #pragma unroll
        for (int ks = 0; ks < 16; ++ks) {
            const int k  = kc + ks * 4;
            const int kk = ks * 4 + koff;
            // A fragments (global, 8B per lane each)
            v2f a0 = *(const v2f*)(xr0 + k);
            v2f a1 = *(const v2f*)(xr1 + k);
            // B fragments: contiguous pair -> single ds_load_b64 each
            v2f b0 = *(const v2f*)(&sWt[c][kk]);
            v2f b1 = *(const v2f*)(&sWt[16 + c][kk]);
            v2f b2 = *(const v2f*)(&sWt[32 + c][kk]);
            v2f b3 = *(const v2f*)(&sWt[48 + c][kk]);
            acc00 = __builtin_amdgcn_wmma_f32_16x16x4_f32(false, a0, false, b0, (short)0, acc00, false, false);
            acc01 = __builtin_amdgcn_wmma_f32_16x16x4_f32(false, a0, false, b1, (short)0, acc01, false, false);
            acc02 = __builtin_amdgcn_wmma_f32_16x16x4_f32(false, a0, false, b2, (short)0, acc02, false, false);
            acc03 = __builtin_amdgcn_wmma_f32_16x16x4_f32(false, a0, false, b3, (short)0, acc03, false, false);
            acc10 = __builtin_amdgcn_wmma_f32_16x16x4_f32(false, a1, false, b0, (short)0, acc10, false, false);
            acc11 = __builtin_amdgcn_wmma_f32_16x16x4_f32(false, a1, false, b1, (short)0, acc11, false, false);
            acc12 = __builtin_amdgcn_wmma_f32_16x16x4_f32(false, a1, false, b2, (short)0, acc12, false, false);
            acc13 = __builtin_amdgcn_wmma_f32_16x16x4_f32(false, a1, false, b3, (short)0, acc13, false, false);
        }
        __syncthreads();
    }

    // D layout: VGPR r -> row = r + 8*(lane>>4), col = lane&15
    const int srow0 = rowBase + 8 * (lane >> 4);
    #pragma unroll
    for (int r = 0; r < 8; ++r) {
        float* g0 = gate + (size_t)(srow0 + r) * NG;
        g0[c]      = acc00[r];
        g0[16 + c] = acc01[r];
        g0[32 + c] = acc02[r];
        g0[48 + c] = acc03[r];
        float* g1 = g0 + (size_t)16 * NG;
        g1[c]      = acc10[r];
        g1[16 + c] = acc11[r];
        g1[32 + c] = acc12[r];
        g1[48 + c] = acc13[r];
    }
}

// ---------------- per-segment max + exp-sum (deterministic) ----------------
// One block (64 threads) per graph; thread t owns gate column t.
__global__ __launch_bounds__(64) void seg_stats_kernel(const float* __restrict__ gate,
                                                       const int* __restrict__ seg_start,
                                                       const int* __restrict__ seg_end,
                                                       float* __restrict__ seg_max,
                                                       float* __restrict__ seg_sum) {
    const int g = blockIdx.x;
    const int t = threadIdx.x;
    const int s = seg_start[g], e = seg_end[g];
    float m = -3.402823466e38f;
    for (int i = s; i < e; ++i) m = fmaxf(m, gate[(size_t)i * NG + t]);
    float sum = 0.0f;
    for (int i = s; i < e; ++i) sum += __expf(gate[(size_t)i * NG + t] - m);
    seg_max[g * NG + t] = m;
    seg_sum[g * NG + t] = sum;
}

// ---------------- per-row mean softmax weight ----------------
// One wave per row; lane handles cols {lane, lane+32}; wave shuffle reduce.
__global__ __launch_bounds__(256) void row_weights_kernel(const float* __restrict__ gate,
                                                          const int* __restrict__ batch,
                                                          const float* __restrict__ seg_max,
                                                          const float* __restrict__ seg_sum,
                                                          float* __restrict__ weights) {
    const int row  = (blockIdx.x * 256 + threadIdx.x) >> 5;
    const int lane = threadIdx.x & 31;
    if (row >= NN) return;
    const int b = batch[row];
    float v = 0.0f;
    #pragma unroll
    for (int h = 0; h < 2; ++h) {
        const int col = lane + 32 * h;
        float e = __expf(gate[(size_t)row * NG + col] - seg_max[b * NG + col]);
        v += e / (seg_sum[b * NG + col] + EPSF);
    }
    #pragma unroll
    for (int off = 16; off > 0; off >>= 1) v += __shfl_xor(v, off, 32);
    if (lane == 0) weights[row] = v * (1.0f / (float)NG);
}

// ---------------- weighted segment pooling (deterministic) ----------------
// One block (256 threads) per graph; thread t owns hidden cols {t, t+256}.
__global__ __launch_bounds__(256) void pool_kernel(const float* __restrict__ x,
                                                   const float* __restrict__ weights,
                                                   const int* __restrict__ seg_start,
                                                   const int* __restrict__ seg_end,
                                                   float* __restrict__ out) {
    const int g = blockIdx.x;
    const int t = threadIdx.x;
    const int s = seg_start[g], e = seg_end[g];
    float a0 = 0.0f, a1 = 0.0f;
    for (int i = s; i < e; ++i) {
        const float w = weights[i];
        a0 += w * x[(size_t)i * HID + t];
        a1 += w * x[(size_t)i * HID + 256 + t];
    }
    out[(size_t)g * HID + t]       = a0;
    out[(size_t)g * HID + 256 + t] = a1;
}

extern "C" void kernel_launch(void* const* d_in, const int* in_sizes, int n_in,
                              void* d_out, int out_size, void* d_ws, size_t ws_size,
                              hipStream_t stream) {
    const float* x     = (const float*)d_in[0];
    const int*   batch = (const int*)d_in[1];
    const float* W     = (const float*)d_in[2];
    float*       out   = (float*)d_out;

    // Workspace layout (bytes)
    char* ws = (char*)d_ws;
    float* gate      = (float*)(ws);                                     // NN*NG*4  = 64 MiB
    float* weights   = (float*)(ws + (size_t)NN * NG * 4);               // NN*4     = 1 MiB
    float* seg_max   = (float*)(ws + (size_t)NN * NG * 4 + (size_t)NN * 4);
    float* seg_sum   = (float*)((char*)seg_max + (size_t)NGRAPH * NG * 4);
    int*   seg_start = (int*)  ((char*)seg_sum + (size_t)NGRAPH * NG * 4);
    int*   seg_end   = (int*)  ((char*)seg_start + (size_t)NGRAPH * 4);

    init_seg_kernel<<<(NGRAPH + 255) / 256, 256, 0, stream>>>(seg_start, seg_end);
    seg_bounds_kernel<<<NN / 256, 256, 0, stream>>>(batch, seg_start, seg_end);
    gate_gemm_kernel<<<NN / 128, 128, 0, stream>>>(x, W, gate);
    seg_stats_kernel<<<NGRAPH, 64, 0, stream>>>(gate, seg_start, seg_end, seg_max, seg_sum);
    row_weights_kernel<<<NN / 8, 256, 0, stream>>>(gate, batch, seg_max, seg_sum, weights);
    pool_kernel<<<NGRAPH, 256, 0, stream>>>(x, weights, seg_start, seg_end, out);
}